// MultiHeadAttention_88845693485593
// MI455X (gfx1250) — compile-verified
//
#include <hip/hip_runtime.h>
#include <math.h>

#define B_ 4
#define S_ 2048
#define D_ 1024
#define H_ 16
#define HD_ 64
#define INV_SCALE 0.125f     // 1/sqrt(HD)
#define LN_EPS 1e-5f

typedef __attribute__((ext_vector_type(2))) float v2f;
typedef __attribute__((ext_vector_type(8))) float v8f;

// D(16x16,f32) = A(16x4,f32) * B(4x16,f32) + C  -- exact-f32 matrix op on CDNA5
__device__ __forceinline__ v8f wmma_f32(v2f a, v2f b, v8f c) {
  return __builtin_amdgcn_wmma_f32_16x16x4_f32(
      /*neg_a=*/false, a, /*neg_b=*/false, b,
      /*c_mod=*/(short)0, c, /*reuse_a=*/false, /*reuse_b=*/false);
}

// CDNA5 async global->LDS copy (16B per lane), tracked by ASYNCcnt.
// LDS destination address = low 32 bits of the generic shared pointer
// (ISA 10.2).
__device__ __forceinline__ void async_copy_b128(const float* gsrc, float* ldst) {
  unsigned int lds_off = (unsigned int)(unsigned long long)ldst;
  asm volatile("global_load_async_to_lds_b128 %0, %1, off"
               :: "v"(lds_off), "v"(gsrc) : "memory");
}
__device__ __forceinline__ void wait_asynccnt_0() {
  asm volatile("s_wait_asynccnt 0x0" ::: "memory");
}
__device__ __forceinline__ void wait_asynccnt_2() {
  asm volatile("s_wait_asynccnt 0x2" ::: "memory");
}

// ---------------------------------------------------------------------------
// GEMM: Y = X[8192,1024] @ W[1024,1024] + bias.
// 4 waves/block; each wave computes a 32x64 tile (8 f32 accumulators).
// W consumed in 16x64 chunks, double-buffered in LDS via async b128 copies.
// head_split=1 scatters output into [B,H,S,HD] layout for attention.
// ---------------------------------------------------------------------------
__global__ __launch_bounds__(128)
void gemm_bias_16x16x4(const float* __restrict__ X, const float* __restrict__ W,
                       const float* __restrict__ bias, float* __restrict__ Y,
                       int head_split) {
  __shared__ __align__(16) float Wt[2][16 * 64];
  const int tid  = threadIdx.x;
  const int lane = tid & 31;
  const int wave = tid >> 5;
  const int m    = lane & 15;      // A row / B,C col within tile
  const int hf   = lane >> 4;      // lane half selects K pair
  const int row0 = blockIdx.x * 128 + wave * 32;
  const int col0 = blockIdx.y * 64;

  // staging assignment: thread copies row skk, cols [sc0, sc0+8) of the chunk
  const int skk = tid >> 3;        // 0..15
  const int sc0 = (tid & 7) * 8;   // 0..56

  {  // prime buffer 0 with chunk 0
    const float* g = W + (size_t)skk * D_ + col0 + sc0;
    float* l = &Wt[0][skk * 64 + sc0];
    async_copy_b128(g, l);
    async_copy_b128(g + 4, l + 4);
  }

  v8f acc[2][4] = {};
  for (int ch = 0; ch < 64; ++ch) {
    const int k0 = ch * 16;
    if (ch < 63) {  // issue next chunk's copy, then wait only for current chunk
      const float* g = W + (size_t)(k0 + 16 + skk) * D_ + col0 + sc0;
      float* l = &Wt[(ch + 1) & 1][skk * 64 + sc0];
      async_copy_b128(g, l);
      async_copy_b128(g + 4, l + 4);
      wait_asynccnt_2();
    } else {
      wait_asynccnt_0();
    }
    __syncthreads();

    const float* Wl = Wt[ch & 1];
#pragma unroll
    for (int kc = 0; kc < 4; ++kc) {
      const int kb = kc * 4 + 2 * hf;   // K offset within chunk
      const float* xr0 = X + (size_t)(row0 + m) * D_ + k0 + kb;
      const float* xr1 = X + (size_t)(row0 + 16 + m) * D_ + k0 + kb;
      v2f a0, a1;
      a0.x = xr0[0]; a0.y = xr0[1];
      a1.x = xr1[0]; a1.y = xr1[1];
#pragma unroll
      for (int t = 0; t < 4; ++t) {
        v2f bb;
        bb.x = Wl[kb * 64 + t * 16 + m];
        bb.y = Wl[(kb + 1) * 64 + t * 16 + m];
        acc[0][t] = wmma_f32(a0, bb, acc[0][t]);
        acc[1][t] = wmma_f32(a1, bb, acc[1][t]);
      }
    }
    __syncthreads();
  }

  const int h = col0 >> 6;   // each 64-col block is exactly one head
#pragma unroll
  for (int i = 0; i < 2; ++i)
#pragma unroll
    for (int t = 0; t < 4; ++t)
#pragma unroll
      for (int r = 0; r < 8; ++r) {
        const int row = row0 + i * 16 + r + 8 * hf;
        const int hd  = t * 16 + m;
        const float val = acc[i][t][r] + bias[col0 + hd];
        if (head_split) {
          const int bi = row >> 11;          // row / S
          const int si = row & (S_ - 1);     // row % S
          Y[(((size_t)bi * H_ + h) * S_ + si) * HD_ + hd] = val;
        } else {
          Y[(size_t)row * D_ + col0 + hd] = val;
        }
      }
}

// ---------------------------------------------------------------------------
// Attention: per (b,h,16-row q-tile), 4 q-tiles per block. K/V tiles are
// contiguous 4KB blocks in [S][HD] layout -> staged once per block via async
// b128 copies and shared by all 4 waves. Multiplicative causal mask: masked
// score := 0 and STILL participates in softmax (exp(0)=1), so we sweep all
// S key tiles with an online softmax. P staged via LDS (C-layout->A-layout).
// ---------------------------------------------------------------------------
__global__ __launch_bounds__(128)
void attention_kernel(const float* __restrict__ Qh, const float* __restrict__ Kh,
                      const float* __restrict__ Vh, float* __restrict__ Ctx) {
  __shared__ __align__(16) float Kl[16 * 64];
  __shared__ __align__(16) float Vl[16 * 64];
  __shared__ __align__(16) float Pbuf[4 * 256];
  const int tid  = threadIdx.x;
  const int lane = tid & 31;
  const int wave = tid >> 5;
  const int m  = lane & 15;
  const int hf = lane >> 4;
  const int b  = blockIdx.z, h = blockIdx.y;
  const int q0 = blockIdx.x * 64 + wave * 16;

  const float* Q = Qh + (size_t)(b * H_ + h) * S_ * HD_;
  const float* K = Kh + (size_t)(b * H_ + h) * S_ * HD_;
  const float* V = Vh + (size_t)(b * H_ + h) * S_ * HD_;
  float* P = Pbuf + wave * 256;

  // Preload this q-row's A-fragments for all 16 K-steps (HD=64).
  v2f qa[16];
#pragma unroll
  for (int s = 0; s < 16; ++s) {
    const int kb = 4 * s + 2 * hf;
    const float* qp = Q + (size_t)(q0 + m) * HD_ + kb;
    qa[s].x = qp[0]; qa[s].y = qp[1];
  }

  v8f   ctxacc[4] = {};
  float rmax[8], rsum[8];
#pragma unroll
  for (int r = 0; r < 8; ++r) { rmax[r] = -1e30f; rsum[r] = 0.f; }

  for (int kt = 0; kt < S_ / 16; ++kt) {
    // stage K/V tiles (4KB each, contiguous) into LDS via async engine
    {
      const float* gK = K + (size_t)kt * 16 * HD_ + tid * 8;
      const float* gV = V + (size_t)kt * 16 * HD_ + tid * 8;
      async_copy_b128(gK,     &Kl[tid * 8]);
      async_copy_b128(gK + 4, &Kl[tid * 8 + 4]);
      async_copy_b128(gV,     &Vl[tid * 8]);
      async_copy_b128(gV + 4, &Vl[tid * 8 + 4]);
      wait_asynccnt_0();
    }
    __syncthreads();

    // scores tile = Qtile(16x64) @ Ktile^T(64x16), K from LDS (b64 reads)
    v8f sc = {};
#pragma unroll
    for (int s = 0; s < 16; ++s) {
      const int kb = 4 * s + 2 * hf;
      const v2f bb = *(const v2f*)&Kl[m * 64 + kb];
      sc = wmma_f32(qa[s], bb, sc);
    }
    // online softmax; shfl masks 1..8 stay inside each 16-lane half = one row
#pragma unroll
    for (int r = 0; r < 8; ++r) {
      const int row = q0 + r + 8 * hf;
      const int col = kt * 16 + m;
      float s = sc[r] * INV_SCALE;
      if (col > row) s = 0.f;               // multiplicative mask: 0, not -inf
      float mx = s;
      mx = fmaxf(mx, __shfl_xor(mx, 1, 32));
      mx = fmaxf(mx, __shfl_xor(mx, 2, 32));
      mx = fmaxf(mx, __shfl_xor(mx, 4, 32));
      mx = fmaxf(mx, __shfl_xor(mx, 8, 32));
      const float nm  = fmaxf(rmax[r], mx);
      const float scl = expf(rmax[r] - nm);
      const float p   = expf(s - nm);
      float ps = p;
      ps += __shfl_xor(ps, 1, 32);
      ps += __shfl_xor(ps, 2, 32);
      ps += __shfl_xor(ps, 4, 32);
      ps += __shfl_xor(ps, 8, 32);
      rsum[r] = rsum[r] * scl + ps;
      rmax[r] = nm;
#pragma unroll
      for (int t = 0; t < 4; ++t) ctxacc[t][r] *= scl;
      P[(r + 8 * hf) * 16 + m] = p;         // stage P in row-major
    }
    __syncthreads();
    // ctx += P(16x16) @ Vtile(16x64), V from LDS
#pragma unroll
    for (int s = 0; s < 4; ++s) {
      const int kb = 4 * s + 2 * hf;
      const v2f a = *(const v2f*)&P[m * 16 + kb];   // A-layout read of P
#pragma unroll
      for (int t = 0; t < 4; ++t) {
        v2f bb;
        bb.x = Vl[kb * 64 + t * 16 + m];
        bb.y = Vl[(kb + 1) * 64 + t * 16 + m];
        ctxacc[t] = wmma_f32(a, bb, ctxacc[t]);
      }
    }
    __syncthreads();   // protect Kl/Vl/P before next tile's staging
  }

#pragma unroll
  for (int t = 0; t < 4; ++t)
#pragma unroll
    for (int r = 0; r < 8; ++r) {
      const int row = q0 + r + 8 * hf;
      const float val = ctxacc[t][r] / rsum[r];
      Ctx[((size_t)b * S_ + row) * D_ + h * HD_ + t * 16 + m] = val;
    }
}

// ---------------------------------------------------------------------------
// LayerNorm(proj + residual) * gamma + beta. One row per block.
// ---------------------------------------------------------------------------
__global__ __launch_bounds__(256)
void ln_residual_kernel(const float* __restrict__ proj, const float* __restrict__ res,
                        const float* __restrict__ gamma, const float* __restrict__ beta,
                        float* __restrict__ out) {
  __shared__ float s1[256];
  __shared__ float s2[256];
  const int row = blockIdx.x;
  const int tid = threadIdx.x;
  float x[4];
  float sum = 0.f, sq = 0.f;
#pragma unroll
  for (int i = 0; i < 4; ++i) {
    const int c = tid + i * 256;
    const float v = proj[(size_t)row * D_ + c] + res[(size_t)row * D_ + c];
    x[i] = v; sum += v; sq += v * v;
  }
  s1[tid] = sum; s2[tid] = sq;
  __syncthreads();
  for (int off = 128; off > 0; off >>= 1) {
    if (tid < off) { s1[tid] += s1[tid + off]; s2[tid] += s2[tid + off]; }
    __syncthreads();
  }
  const float mean = s1[0] * (1.f / D_);
  const float var  = s2[0] * (1.f / D_) - mean * mean;
  const float rstd = rsqrtf(var + LN_EPS);
#pragma unroll
  for (int i = 0; i < 4; ++i) {
    const int c = tid + i * 256;
    out[(size_t)row * D_ + c] = (x[i] - mean) * rstd * gamma[c] + beta[c];
  }
}

extern "C" void kernel_launch(void* const* d_in, const int* in_sizes, int n_in,
                              void* d_out, int out_size, void* d_ws, size_t ws_size,
                              hipStream_t stream) {
  (void)in_sizes; (void)n_in; (void)out_size; (void)ws_size;
  const float* q     = (const float*)d_in[0];
  const float* k     = (const float*)d_in[1];
  const float* v     = (const float*)d_in[2];
  const float* Wq    = (const float*)d_in[3];
  const float* bq    = (const float*)d_in[4];
  const float* Wk    = (const float*)d_in[5];
  const float* bk    = (const float*)d_in[6];
  const float* Wv    = (const float*)d_in[7];
  const float* bv    = (const float*)d_in[8];
  const float* Wp    = (const float*)d_in[9];
  const float* bp    = (const float*)d_in[10];
  const float* gamma = (const float*)d_in[11];
  const float* beta  = (const float*)d_in[12];
  float* out = (float*)d_out;
  float* ws  = (float*)d_ws;

  const size_t HSZ = (size_t)B_ * H_ * S_ * HD_;   // 8M floats
  float* qh   = ws;
  float* kh   = ws + HSZ;
  float* vh   = ws + 2 * HSZ;
  float* ctx  = ws + 3 * HSZ;                       // [B,S,D]
  float* proj = ws;                                 // reuse qh after attention

  const dim3 gblk(128, 1, 1);
  const dim3 ggrid((B_ * S_) / 128, D_ / 64, 1);
  gemm_bias_16x16x4<<<ggrid, gblk, 0, stream>>>(q, Wq, bq, qh, 1);
  gemm_bias_16x16x4<<<ggrid, gblk, 0, stream>>>(k, Wk, bk, kh, 1);
  gemm_bias_16x16x4<<<ggrid, gblk, 0, stream>>>(v, Wv, bv, vh, 1);

  const dim3 agrid(S_ / 64, H_, B_);
  attention_kernel<<<agrid, dim3(128, 1, 1), 0, stream>>>(qh, kh, vh, ctx);

  gemm_bias_16x16x4<<<ggrid, gblk, 0, stream>>>(ctx, Wp, bp, proj, 0);

  ln_residual_kernel<<<dim3(B_ * S_, 1, 1), dim3(256, 1, 1), 0, stream>>>(
      proj, q, gamma, beta, out);
}